// UpFu_77464030151216
// MI455X (gfx1250) — compile-verified
//
#include <hip/hip_runtime.h>
#include <hip/hip_bf16.h>
#include <math.h>

// ---------------------------------------------------------------------------
// CDNA5 fp32 WMMA: D(16x16) = A(16x4) * B(4x16) + C, full fp32 precision.
// ---------------------------------------------------------------------------
typedef float v2f __attribute__((ext_vector_type(2)));
typedef float v8f __attribute__((ext_vector_type(8)));
typedef unsigned int v4u __attribute__((ext_vector_type(4)));
typedef int v8i __attribute__((ext_vector_type(8)));
typedef int v4i __attribute__((ext_vector_type(4)));

__device__ __forceinline__ v8f wmma4(v2f a, v2f b, v8f c) {
  // 8 args: (neg_a, A, neg_b, B, c_mod, C, reuse_a, reuse_b)
  return __builtin_amdgcn_wmma_f32_16x16x4_f32(false, a, false, b, (short)0, c,
                                               false, false);
}

// ---------------------------------------------------------------------------
// TDM: contiguous 1-D DMA global -> LDS (Tensor Data Mover, ISA ch.8 D#).
// group0: [count=1 | lds_addr | global_addr[31:0] | ga[56:32] + type=2]
// group1: [data_size=4B | tensor_dim0=n | tensor_dim1=1 | tile_dim0=n | ...]
// This toolchain exposes the 6-arg builtin: (g0, g1, g2, g3, g4, cpol).
// Issue from ONE wave only (EXEC is ignored; one DMA per executing wave).
// ---------------------------------------------------------------------------
__device__ __forceinline__ void tdm_copy_1d(const float* gsrc, float* ldsdst,
                                            unsigned nelem) {
  unsigned long long ga = (unsigned long long)(uintptr_t)gsrc;
  unsigned lds = (unsigned)(uintptr_t)ldsdst;   // low 32 bits = LDS offset
  v4u g0;
  g0.x = 1u;                                          // count=1, user mode
  g0.y = lds;                                         // lds_addr (bytes)
  g0.z = (unsigned)ga;                                // global_addr[31:0]
  g0.w = (unsigned)((ga >> 32) & 0x01FFFFFFu) | (2u << 30);  // ga[56:32]|type=2
  v8i g1;
  g1[0] = (int)(2u << 16);                            // data_size=2 (4 bytes)
  g1[1] = (int)((nelem & 0xFFFFu) << 16);             // tensor_dim0 lo16
  g1[2] = (int)(((nelem >> 16) & 0xFFFFu) | (1u << 16)); // dim0 hi | dim1=1
  g1[3] = (int)((nelem & 0xFFFFu) << 16);             // tile_dim0 = nelem
  g1[4] = 0;                                          // tile_dim1/2 unused
  g1[5] = (int)nelem;                                 // dim0_stride lo32
  g1[6] = 0;
  g1[7] = 0;
  v4i zz = {0, 0, 0, 0};                              // <=2D: groups 2/3 NULL
  v8i z8 = {0, 0, 0, 0, 0, 0, 0, 0};
  __builtin_amdgcn_tensor_load_to_lds(g0, g1, zz, zz, z8, 0);
}

// ---------------------------------------------------------------------------
// Problem constants
// ---------------------------------------------------------------------------
#define NB   8
#define CH   256
#define CL   128
#define HW   128
#define HW2  16384
#define SH   64
#define SH2  4096

// ---------------------------------------------------------------------------
// 0) One-shot weight repacks so TDM stages are contiguous and B fragments are
//    aligned v2f pairs:  wT1[tap][k/2][n][2],  pT[k/2][n][2]
// ---------------------------------------------------------------------------
__global__ __launch_bounds__(256) void k_wt1(const float* __restrict__ w1,
                                             float* __restrict__ wT) {
  int gid = blockIdx.x * 256 + threadIdx.x;          // 9*384*64 = 221184
  int j = gid & 1;
  int n = (gid >> 1) & 63;
  int k2 = (gid >> 7) % 192;
  int tap = gid / 24576;                              // 192*64*2
  int k = k2 * 2 + j;
  wT[gid] = w1[(size_t)n * 3456 + (size_t)k * 9 + tap];
}

__global__ __launch_bounds__(256) void k_wt2(const float* __restrict__ pw,
                                             float* __restrict__ pT) {
  int gid = blockIdx.x * 256 + threadIdx.x;          // 128*128*2 = 32768
  int j = gid & 1;
  int n = (gid >> 1) & 127;
  int k2 = gid >> 8;
  pT[gid] = pw[(size_t)n * 256 + k2 * 2 + j];
}

// ---------------------------------------------------------------------------
// 1) Bilinear 2x upsample of f_high -> fh_r (half-pixel centers, edge clamp)
// ---------------------------------------------------------------------------
__global__ __launch_bounds__(256) void k_resize(const float* __restrict__ fh,
                                                float* __restrict__ out) {
  int gid = blockIdx.x * 256 + threadIdx.x;          // NB*CH*HW2
  int bc = gid >> 14;
  int pp = gid & (HW2 - 1);
  int y = pp >> 7, x = pp & 127;
  float xi = x * 0.5f - 0.25f, yi = y * 0.5f - 0.25f;
  float xf = floorf(xi), yf = floorf(yi);
  float fx = xi - xf, fy = yi - yf;
  int xfi = (int)xf, yfi = (int)yf;
  int x0 = min(max(xfi, 0), SH - 1), x1 = min(max(xfi + 1, 0), SH - 1);
  int y0 = min(max(yfi, 0), SH - 1), y1 = min(max(yfi + 1, 0), SH - 1);
  const float* p = fh + (size_t)bc * SH2;
  float v00 = p[y0 * SH + x0], v01 = p[y0 * SH + x1];
  float v10 = p[y1 * SH + x0], v11 = p[y1 * SH + x1];
  out[gid] = (1.f - fy) * ((1.f - fx) * v00 + fx * v01) +
             fy * ((1.f - fx) * v10 + fx * v11);
}

// ---------------------------------------------------------------------------
// 2) conv1 3x3 (384 -> 64) + ReLU as implicit GEMM via fp32 WMMA.
//    Block = one (b,y) row; 8 waves x 16-pixel M tiles; weights staged per
//    (tap,chunk) by TDM into 48KB LDS; branch-free padded A loads.
// ---------------------------------------------------------------------------
#define KC1 192
__global__ __launch_bounds__(256) void k_conv1(const float* __restrict__ fhr,
                                               const float* __restrict__ flow,
                                               const float* __restrict__ wT,
                                               const float* __restrict__ b1,
                                               float* __restrict__ hid) {
  __shared__ float wL[KC1 * 64];                     // 48 KB, [k/2][n][2]
  int blk = blockIdx.x;                              // b*128 + y
  int b = blk >> 7, y = blk & 127;
  int tid = threadIdx.x;
  int wv = tid >> 5, lane = tid & 31;
  int x0 = wv * 16;
  int n = lane & 15;
  int klane = (lane >> 4) * 2;
  v8f acc[4] = {};
  const float* fhbase = fhr + (size_t)b * CH * HW2;
  const float* flbase = flow + (size_t)b * CL * HW2;

  for (int tap = 0; tap < 9; ++tap) {
    int ky = tap / 3 - 1, kx = tap % 3 - 1;
    int yy = y + ky;
    if (yy < 0 || yy > 127) continue;                // block-uniform skip
    int xx = x0 + n + kx;
    float am = (xx >= 0 && xx <= 127) ? 1.f : 0.f;   // branch-free zero-pad
    int xxc = min(max(xx, 0), 127);
    const float* arow_h = fhbase + yy * HW + xxc;
    const float* arow_l = flbase + yy * HW + xxc;
    for (int chunk = 0; chunk < 2; ++chunk) {
      int cbase = chunk * KC1;
      __syncthreads();
      if (wv == 0) {                                 // one DMA per stage
        tdm_copy_1d(wT + ((size_t)tap * 192 + (cbase >> 1)) * 128, wL,
                    KC1 * 64);
        __builtin_amdgcn_s_wait_tensorcnt(0);
      }
      __syncthreads();
      for (int cs = 0; cs < KC1 / 4; ++cs) {
        int c0 = cbase + cs * 4 + klane;
        const float* src = (c0 < CH)                  // wave-uniform branch
                               ? arow_h + (size_t)c0 * HW2
                               : arow_l + (size_t)(c0 - CH) * HW2;
        v2f a;
        a.x = am * src[0];
        a.y = am * src[HW2];
        int kb = cs * 4 + klane;
#pragma unroll
        for (int t = 0; t < 4; ++t) {
          v2f bb = *(const v2f*)(wL + ((kb >> 1) << 7) + ((t * 16 + n) << 1));
          acc[t] = wmma4(a, bb, acc[t]);
        }
      }
    }
  }
  int rowsel = (lane >> 4) * 8;
#pragma unroll
  for (int t = 0; t < 4; ++t) {
    int co = t * 16 + n;
    float bias = b1[co];
#pragma unroll
    for (int r = 0; r < 8; ++r) {
      int xo = x0 + r + rowsel;
      float v = acc[t][r] + bias;
      hid[((size_t)(b * 64 + co) * HW + y) * HW + xo] = fmaxf(v, 0.f);
    }
  }
}

// ---------------------------------------------------------------------------
// 3) conv2 3x3 (64 -> 2): offsets. Small; plain VALU.
// ---------------------------------------------------------------------------
__global__ __launch_bounds__(256) void k_conv2(const float* __restrict__ hid,
                                               const float* __restrict__ w2,
                                               const float* __restrict__ b2,
                                               float* __restrict__ off) {
  int gid = blockIdx.x * 256 + threadIdx.x;          // NB*2*HW2
  int b = gid >> 15;
  int o = (gid >> 14) & 1;
  int pp = gid & (HW2 - 1);
  int y = pp >> 7, x = pp & 127;
  float acc = b2[o];
  const float* hb = hid + (size_t)b * 64 * HW2;
  for (int ci = 0; ci < 64; ++ci) {
    const float* hc = hb + (size_t)ci * HW2;
    const float* wc = w2 + o * 576 + ci * 9;
#pragma unroll
    for (int ky = 0; ky < 3; ++ky) {
      int yy = y + ky - 1;
      if (yy < 0 || yy > 127) continue;
#pragma unroll
      for (int kx = 0; kx < 3; ++kx) {
        int xx = x + kx - 1;
        if (xx < 0 || xx > 127) continue;
        acc += wc[ky * 3 + kx] * hc[yy * HW + xx];
      }
    }
  }
  off[gid] = acc;
}

// ---------------------------------------------------------------------------
// 4) Fused grid-sample (border clamp) + 1x1 projection GEMM via WMMA.
//    proj_w staged via TDM in 32KB chunks; deterministic partial sums for p.
// ---------------------------------------------------------------------------
#define KC2 64
__global__ __launch_bounds__(256) void k_gproj(const float* __restrict__ fh,
                                               const float* __restrict__ off,
                                               const float* __restrict__ pT,
                                               const float* __restrict__ pb,
                                               float* __restrict__ X,
                                               float* __restrict__ partial) {
  __shared__ float pL[KC2 * 128];                    // 32 KB, [k/2][n][2]
  __shared__ float psum[8 * 128];                    // 4 KB
  int blk = blockIdx.x;                              // 1024, 128 per batch
  int b = blk >> 7;
  int p0 = (blk & 127) * 128;
  int tid = threadIdx.x, wv = tid >> 5, lane = tid & 31;
  int n = lane & 15, klane = (lane >> 4) * 2;

  // Per-lane sampling params for pixel (p0 + wv*16 + (lane&15)).
  int pix = p0 + wv * 16 + n;
  int y = pix >> 7, x = pix & 127;
  float offx = off[(size_t)(b * 2) * HW2 + pix];
  float offy = off[(size_t)(b * 2 + 1) * HW2 + pix];
  float gx = (2.0f / 127.0f) * x - 1.0f + offx * (2.0f / 128.0f);
  float gy = (2.0f / 127.0f) * y - 1.0f + offy * (2.0f / 128.0f);
  float ix = fminf(fmaxf(((gx + 1.f) * 64.f - 1.f) * 0.5f, 0.f), 63.f);
  float iy = fminf(fmaxf(((gy + 1.f) * 64.f - 1.f) * 0.5f, 0.f), 63.f);
  float xf = floorf(ix), yf = floorf(iy);
  float wx = ix - xf, wy = iy - yf;
  int x0 = min(max((int)xf, 0), 63), y0 = min(max((int)yf, 0), 63);
  int x1 = min(x0 + 1, 63), y1 = min(y0 + 1, 63);
  int o00 = y0 * SH + x0, o01 = y0 * SH + x1;
  int o10 = y1 * SH + x0, o11 = y1 * SH + x1;
  float w00 = (1.f - wx) * (1.f - wy), w01 = wx * (1.f - wy);
  float w10 = (1.f - wx) * wy, w11 = wx * wy;
  const float* fb = fh + (size_t)b * CH * SH2;

  v8f acc[8] = {};
  for (int chunk = 0; chunk < 4; ++chunk) {
    int cb = chunk * KC2;
    __syncthreads();
    if (wv == 0) {
      tdm_copy_1d(pT + (size_t)chunk * (KC2 / 2) * 256, pL, (KC2 / 2) * 256);
      __builtin_amdgcn_s_wait_tensorcnt(0);
    }
    __syncthreads();
    for (int cs = 0; cs < KC2 / 4; ++cs) {
      int c0 = cb + cs * 4 + klane;
      const float* p0c = fb + (size_t)c0 * SH2;
      const float* p1c = p0c + SH2;
      v2f a;
      a.x = w00 * p0c[o00] + w01 * p0c[o01] + w10 * p0c[o10] + w11 * p0c[o11];
      a.y = w00 * p1c[o00] + w01 * p1c[o01] + w10 * p1c[o10] + w11 * p1c[o11];
      int kb = cs * 4 + klane;
#pragma unroll
      for (int t = 0; t < 8; ++t) {
        v2f bb = *(const v2f*)(pL + ((kb >> 1) << 8) + ((t * 16 + n) << 1));
        acc[t] = wmma4(a, bb, acc[t]);
      }
    }
  }

  int rowsel = (lane >> 4) * 8;
#pragma unroll
  for (int t = 0; t < 8; ++t) {
    int co = t * 16 + n;
    float bias = pb[co];
    float s = 0.f;
#pragma unroll
    for (int r = 0; r < 8; ++r) {
      int pox = p0 + wv * 16 + r + rowsel;
      float v = acc[t][r] + bias;
      X[((size_t)b * 128 + co) * HW2 + pox] = v;
      s += v;
    }
    s += __shfl_xor(s, 16);                          // fold the two halves
    if (lane < 16) psum[wv * 128 + co] = s;
  }
  __syncthreads();
  if (tid < 128) {                                   // deterministic block sum
    float s = 0.f;
    for (int w8 = 0; w8 < 8; ++w8) s += psum[w8 * 128 + tid];
    partial[(size_t)blk * 128 + tid] = s;
  }
}

// 4b) reduce per-block partials -> p (spatial mean per (b,co)), deterministic
__global__ __launch_bounds__(256) void k_pmean(const float* __restrict__ partial,
                                               float* __restrict__ p) {
  int g = blockIdx.x * 256 + threadIdx.x;            // 1024
  if (g >= 1024) return;
  int b = g >> 7, co = g & 127;
  float s = 0.f;
  for (int blk = 0; blk < 128; ++blk)
    s += partial[((size_t)(b * 128 + blk)) * 128 + co];
  p[g] = s * (1.0f / 16384.0f);
}

// ---------------------------------------------------------------------------
// 5) channel attention: a = sigmoid(relu(p@W1^T+b1)@W2^T+b2)
// ---------------------------------------------------------------------------
__global__ __launch_bounds__(128) void k_ca(const float* __restrict__ p,
                                            const float* __restrict__ w1,
                                            const float* __restrict__ b1,
                                            const float* __restrict__ w2,
                                            const float* __restrict__ b2,
                                            float* __restrict__ a) {
  __shared__ float ps[128], hs[16];
  int b = blockIdx.x, t = threadIdx.x;
  ps[t] = p[b * 128 + t];
  __syncthreads();
  if (t < 16) {
    float h = b1[t];
    for (int c = 0; c < 128; ++c) h += ps[c] * w1[t * 128 + c];
    hs[t] = fmaxf(h, 0.f);
  }
  __syncthreads();
  float v = b2[t];
  for (int j = 0; j < 16; ++j) v += hs[j] * w2[t * 16 + j];
  a[b * 128 + t] = 1.f / (1.f + __expf(-v));
}

// ---------------------------------------------------------------------------
// 6) sm = [mean_c(x*a), max_c(x*a)]
// ---------------------------------------------------------------------------
__global__ __launch_bounds__(256) void k_sm(const float* __restrict__ X,
                                            const float* __restrict__ a,
                                            float* __restrict__ sm) {
  int gid = blockIdx.x * 256 + threadIdx.x;          // NB*HW2
  int b = gid >> 14, pp = gid & (HW2 - 1);
  const float* xb = X + (size_t)b * 128 * HW2 + pp;
  const float* ab = a + b * 128;
  float s = 0.f, mx = -3.402823466e38f;
  for (int c = 0; c < 128; ++c) {
    __builtin_prefetch(xb + (size_t)(c + 8) * HW2, 0, 1);
    float v = xb[(size_t)c * HW2] * ab[c];
    s += v;
    mx = fmaxf(mx, v);
  }
  sm[(size_t)(b * 2) * HW2 + pp] = s * (1.f / 128.f);
  sm[(size_t)(b * 2 + 1) * HW2 + pp] = mx;
}

// ---------------------------------------------------------------------------
// 7) sa = sigmoid(conv7x7(sm)), pad=3, zero padding
// ---------------------------------------------------------------------------
__global__ __launch_bounds__(256) void k_sa(const float* __restrict__ sm,
                                            const float* __restrict__ w,
                                            const float* __restrict__ bb,
                                            float* __restrict__ sa) {
  int gid = blockIdx.x * 256 + threadIdx.x;          // NB*HW2
  int b = gid >> 14, pp = gid & (HW2 - 1);
  int y = pp >> 7, x = pp & 127;
  float acc = bb[0];
  for (int ch = 0; ch < 2; ++ch) {
    const float* s = sm + (size_t)(b * 2 + ch) * HW2;
    const float* wc = w + ch * 49;
    for (int ky = 0; ky < 7; ++ky) {
      int yy = y + ky - 3;
      if (yy < 0 || yy > 127) continue;
      for (int kx = 0; kx < 7; ++kx) {
        int xx = x + kx - 3;
        if (xx < 0 || xx > 127) continue;
        acc += wc[ky * 7 + kx] * s[yy * HW + xx];
      }
    }
  }
  sa[gid] = 1.f / (1.f + __expf(-acc));
}

// ---------------------------------------------------------------------------
// 8) out = x * a * sa
// ---------------------------------------------------------------------------
__global__ __launch_bounds__(256) void k_out(const float* __restrict__ X,
                                             const float* __restrict__ a,
                                             const float* __restrict__ sa,
                                             float* __restrict__ out) {
  size_t gid = (size_t)blockIdx.x * 256 + threadIdx.x;  // NB*128*HW2
  int b = (int)(gid >> 21);
  int co = (int)(gid >> 14) & 127;
  int pp = (int)gid & (HW2 - 1);
  out[gid] = X[gid] * a[b * 128 + co] * sa[(size_t)b * HW2 + pp];
}

// ---------------------------------------------------------------------------
// Launch
// ---------------------------------------------------------------------------
extern "C" void kernel_launch(void* const* d_in, const int* in_sizes, int n_in,
                              void* d_out, int out_size, void* d_ws,
                              size_t ws_size, hipStream_t stream) {
  const float* f_high = (const float*)d_in[0];
  const float* f_low  = (const float*)d_in[1];
  const float* off_w1 = (const float*)d_in[2];
  const float* off_b1 = (const float*)d_in[3];
  const float* off_w2 = (const float*)d_in[4];
  const float* off_b2 = (const float*)d_in[5];
  const float* proj_w = (const float*)d_in[6];
  const float* proj_b = (const float*)d_in[7];
  const float* ca_w1  = (const float*)d_in[8];
  const float* ca_b1  = (const float*)d_in[9];
  const float* ca_w2  = (const float*)d_in[10];
  const float* ca_b2  = (const float*)d_in[11];
  const float* sa_w   = (const float*)d_in[12];
  const float* sa_b   = (const float*)d_in[13];
  float* out = (float*)d_out;

  // Workspace carve (floats):
  float* fhr     = (float*)d_ws;                 // 33,554,432
  float* hid     = fhr + (size_t)NB * CH * HW2;  //  8,388,608
  float* offb    = hid + (size_t)NB * 64 * HW2;  //    262,144
  float* X       = offb + (size_t)NB * 2 * HW2;  // 16,777,216
  float* partial = X + (size_t)NB * 128 * HW2;   //    131,072
  float* p       = partial + 1024 * 128;         //      1,024
  float* av      = p + 1024;                     //      1,024
  float* sm      = av + 1024;                    //    262,144
  float* sa      = sm + (size_t)NB * 2 * HW2;    //    131,072
  float* wT1     = sa + (size_t)NB * HW2;        //    221,184
  float* pT      = wT1 + 9 * 384 * 64;           //     32,768

  k_wt1<<<(9 * 384 * 64) / 256, 256, 0, stream>>>(off_w1, wT1);
  k_wt2<<<(128 * 128 * 2) / 256, 256, 0, stream>>>(proj_w, pT);
  k_resize<<<(NB * CH * HW2) / 256, 256, 0, stream>>>(f_high, fhr);
  k_conv1<<<NB * HW, 256, 0, stream>>>(fhr, f_low, wT1, off_b1, hid);
  k_conv2<<<(NB * 2 * HW2) / 256, 256, 0, stream>>>(hid, off_w2, off_b2, offb);
  k_gproj<<<NB * 128, 256, 0, stream>>>(f_high, offb, pT, proj_b, X, partial);
  k_pmean<<<4, 256, 0, stream>>>(partial, p);
  k_ca<<<NB, 128, 0, stream>>>(p, ca_w1, ca_b1, ca_w2, ca_b2, av);
  k_sm<<<(NB * HW2) / 256, 256, 0, stream>>>(X, av, sm);
  k_sa<<<(NB * HW2) / 256, 256, 0, stream>>>(sm, sa_w, sa_b, sa);
  k_out<<<(NB * 128 * HW2) / 256, 256, 0, stream>>>(X, av, sa, out);
}